// BIODecoder_41970420417655
// MI455X (gfx1250) — compile-verified
//
#include <hip/hip_runtime.h>

// ---------------------------------------------------------------------------
// Linear-chain CRF forward (BIO-constrained), mean NLL, for MI455X (gfx1250).
//
// Kernel 1 (sequential scan, WMMA): rescaled linear-domain forward scan.
//   Per 16-batch tile and per time step:
//     Q_new = exp(em_t) .* (E^T x Q_old)     (elementwise exp on D layout)
//   E = exp(masked transitions) is CONSTANT -> A operand of
//   V_WMMA_F32_16X16X4_F32 held in 6 VGPRs for the whole scan (3 K-slices).
//   Q lives in the WMMA C/D layout (rows = tag in VGPR idx + half-wave,
//   cols = batch = lane&15); D->B conversion per step = 6 ds_swizzle
//   (SWAPX16) half-swaps + 6 cndmasks.  Per-batch rescale (log2 domain,
//   raw v_exp/v_log/v_rcp) keeps f32 in range; the (q,c) state is
//   snapshotted branchlessly at t == len-1 and Z is computed once at end.
//
//   NOTE: emissions are kept in TWO register arrays (eLo[8], eHi[3]) so the
//   half-wave select  lo ? eLo[j] : eHi[j]  can never be canonicalized into
//   a dynamically-indexed register gather (select waterfall).
//
// Kernel 2 (gold-path score): parallel over (b,t), kept out of the scan.
// Kernel 3: deterministic tree reduction -> mean NLL.
// ---------------------------------------------------------------------------

#define TT 512
#define BBATCH 4096
#define KK 11
#define IMPOSSIBLE_F (-10000.0f)
#define LOG2E_F 1.44269504088896340736f
#define LN2_F 0.69314718055994530942f

typedef float v2f __attribute__((ext_vector_type(2)));
typedef float v4f __attribute__((ext_vector_type(4), aligned(4)));
typedef float v8f __attribute__((ext_vector_type(8)));

// cross-lane half-wave swap (lanes l <-> l^16), wave32 group-of-32 swizzle
#define SWZF(x, imm) __int_as_float(__builtin_amdgcn_ds_swizzle(__float_as_int(x), (imm)))
__device__ __forceinline__ float swap16(float x) { return SWZF(x, 0x401F); }

__device__ __forceinline__ float exp2r(float x) { return __builtin_amdgcn_exp2f(x); }
__device__ __forceinline__ float log2r(float x) { return __builtin_amdgcn_logf(x); }
__device__ __forceinline__ float expn(float x) { return exp2r(x * LOG2E_F); }

__device__ __forceinline__ void loadEm(float* eLo, float* eHi, const float* base, int t) {
  const char* p = (const char*)(base + (size_t)t * KK);
  v4f x = *(const v4f*)(p);        // em[0..3]
  v4f y = *(const v4f*)(p + 16);   // em[4..7]
  v4f w = *(const v4f*)(p + 28);   // em[7..10] (overlapped read, in-bounds)
  eLo[0] = x[0]; eLo[1] = x[1]; eLo[2] = x[2]; eLo[3] = x[3];
  eLo[4] = y[0]; eLo[5] = y[1]; eLo[6] = y[2]; eLo[7] = y[3];
  eHi[0] = w[1]; eHi[1] = w[2]; eHi[2] = w[3];
}

// ============================ Kernel 1: scan ===============================
__global__ void __launch_bounds__(32)
crf_scan(const float* __restrict__ em, const int* __restrict__ mask,
         const float* __restrict__ trans, const float* __restrict__ startT,
         const float* __restrict__ endT, float* __restrict__ zOut) {
  __shared__ float sTrans[KK * KK];
  __shared__ float sStart[KK];
  __shared__ float sEnd[KK];

  // BIO-masked parameters into LDS.
  // allowed(k -> j): j==O, or j is a B-tag, or j is I_i and k in {B_i, I_i}
  for (int i = threadIdx.x; i < KK * KK; i += 32) {
    int k = i / KK, j = i % KK;
    bool allowed = (j == 0) || ((j & 1) == 1) || (k == j) || (k == j - 1);
    sTrans[i] = allowed ? trans[i] : IMPOSSIBLE_F;
  }
  if (threadIdx.x < KK) {
    int i = threadIdx.x;
    sStart[i] = ((i >= 2) && ((i & 1) == 0)) ? IMPOSSIBLE_F : startT[i];
    sEnd[i] = endT[i];
  }
  __syncthreads();

  const int lane = threadIdx.x & 31;
  const bool lo = lane < 16;
  const int mrow = lane & 15;                 // batch column / A-matrix M row
  const int b = blockIdx.x * 16 + mrow;       // one 16-batch tile per wave
  const float* emB = em + (size_t)b * TT * KK;
  const int* mkRow = mask + (size_t)b * TT;

  // sequence length: branchless binary search on the prefix mask
  int blo = 0, bhi = TT;
  for (int it = 0; it < 10; ++it) {
    int mid = (blo + bhi) >> 1;
    int mv = mkRow[mid];
    bool active = blo < bhi;
    bool take = active && (mv != 0);
    blo = take ? (mid + 1) : blo;
    bhi = (active && !take) ? mid : bhi;
  }
  const int len = blo;

  // constant A operand: E^T = exp(masked trans)^T, WMMA A layout
  // slice s covers K = 4s..4s+3 : VGPR0 = K=4s (lo) / 4s+2 (hi),
  //                               VGPR1 = K=4s+1 (lo) / 4s+3 (hi)
  v2f A0, A1, A2;
  {
    const int i = mrow;  // M row
#pragma unroll
    for (int s = 0; s < 3; ++s) {
      int k0 = 4 * s + (lo ? 0 : 2);
      int k1 = k0 + 1;
      int c0 = (k0 < KK) ? k0 : 0, c1 = (k1 < KK) ? k1 : 0;
      float a0v = (k0 < KK && i < KK) ? expn(sTrans[c0 * KK + i]) : 0.0f;
      float a1v = (k1 < KK && i < KK) ? expn(sTrans[c1 * KK + i]) : 0.0f;
      if (s == 0) { A0[0] = a0v; A0[1] = a1v; }
      else if (s == 1) { A1[0] = a0v; A1[1] = a1v; }
      else { A2[0] = a0v; A2[1] = a1v; }
    }
  }

  // exp(end) in the D layout (per-lane row constants)
  float ee[8];
#pragma unroll
  for (int j = 0; j < 8; ++j) {
    int tg_ = lo ? j : (j + 8);
    ee[j] = (tg_ < KK) ? expn(sEnd[(tg_ < KK) ? tg_ : 0]) : 0.0f;
  }

  // ---- init state at t = 0 (c kept in log2 domain) ----
  float c2;
  v8f q;
  {
    float e0Lo[8], e0Hi[3];
    loadEm(e0Lo, e0Hi, emB, 0);
    float v[8];
#pragma unroll
    for (int j = 0; j < 8; ++j) {
      float valLo = sStart[j] + e0Lo[j];
      float valHi = (j < 3) ? (sStart[(j < 3) ? (j + 8) : 0] + e0Hi[(j < 3) ? j : 0])
                            : -1.0e30f;
      float val = lo ? valLo : valHi;
      v[j] = val * LOG2E_F;
    }
    float m = v[0];
#pragma unroll
    for (int j = 1; j < 8; ++j) m = fmaxf(m, v[j]);
    m = fmaxf(m, swap16(m));
    c2 = m;
#pragma unroll
    for (int j = 0; j < 8; ++j) q[j] = exp2r(v[j] - m);
  }
  // Z snapshot state; seeding with the t=0 state also covers len <= 1.
  v8f qs = q;
  float cs = c2;

  // ---- main sequential scan over T, double-buffered loads ----
  float eLoC[8], eHiC[3];
  loadEm(eLoC, eHiC, emB, 1);

#pragma unroll 2
  for (int t = 1; t < TT; ++t) {
    // stage next-step emissions + deep prefetch (HBM latency hiding)
    int tn = (t + 1 < TT) ? (t + 1) : t;
    float eLoN[8], eHiN[3];
    loadEm(eLoN, eHiN, emB, tn);
    int tp = (t + 8 < TT) ? (t + 8) : (TT - 1);
    __builtin_prefetch(emB + (size_t)tp * KK, 0, 3);  // global_prefetch_b8

    // B operand (K-slices of Q) via half-wave swizzle swaps
    v2f B0, B1, B2;
    B0[0] = lo ? q[0] : swap16(q[2]);  // rows 0 / 2
    B0[1] = lo ? q[1] : swap16(q[3]);  // rows 1 / 3
    B1[0] = lo ? q[4] : swap16(q[6]);  // rows 4 / 6
    B1[1] = lo ? q[5] : swap16(q[7]);  // rows 5 / 7
    B2[0] = lo ? swap16(q[0]) : q[2];  // rows 8 / 10
    B2[1] = lo ? swap16(q[1]) : 0.0f;  // rows 9 / pad

    // D = E^T x Q : three chained f32 WMMAs
    v8f d = {};
    d = __builtin_amdgcn_wmma_f32_16x16x4_f32(false, A0, false, B0, (short)0, d, false, false);
    d = __builtin_amdgcn_wmma_f32_16x16x4_f32(false, A1, false, B1, (short)0, d, false, false);
    d = __builtin_amdgcn_wmma_f32_16x16x4_f32(false, A2, false, B2, (short)0, d, false, false);

    // elementwise exp(emissions) in D layout.
    // eLo/eHi are DISTINCT arrays -> value-select, never an index-select.
#pragma unroll
    for (int j = 0; j < 8; ++j) {
      float src = (j < 3) ? (lo ? eLoC[j] : eHiC[(j < 3) ? j : 0]) : eLoC[j];
      float ex = exp2r(src * LOG2E_F);
      float f = (j < 3) ? ex : (lo ? ex : 0.0f);  // pad rows stay zero
      q[j] = d[j] * f;
    }

    // per-batch rescale, accumulate log2-scale
    float m = q[0];
#pragma unroll
    for (int j = 1; j < 8; ++j) m = fmaxf(m, q[j]);
    m = fmaxf(m, swap16(m));
    float r = __builtin_amdgcn_rcpf(m);
    c2 += log2r(m);
#pragma unroll
    for (int j = 0; j < 8; ++j) q[j] *= r;

    // branchless state snapshot at t == len-1
    bool hit = (t == len - 1);
#pragma unroll
    for (int j = 0; j < 8; ++j) qs[j] = hit ? q[j] : qs[j];
    cs = hit ? c2 : cs;

#pragma unroll
    for (int j = 0; j < 8; ++j) eLoC[j] = eLoN[j];
#pragma unroll
    for (int j = 0; j < 3; ++j) eHiC[j] = eHiN[j];
  }

  // Z from the snapshotted state (log2 -> natural log once)
  float dot = 0.0f;
#pragma unroll
  for (int j = 0; j < 8; ++j) dot = fmaf(qs[j], ee[j], dot);
  dot += swap16(dot);
  float z = (cs + log2r(dot)) * LN2_F;

  if (lo) zOut[blockIdx.x * 16 + mrow] = z;  // halves hold identical z
}

// ===================== Kernel 2: gold-path score ===========================
// One block per sequence; parallel over t; nll[b] = score[b] - z[b] in place.
__global__ void __launch_bounds__(256)
crf_score(const float* __restrict__ em, const int* __restrict__ mask,
          const int* __restrict__ tags, const float* __restrict__ trans,
          const float* __restrict__ startT, const float* __restrict__ endT,
          float* __restrict__ znll) {
  __shared__ float sTrans[KK * KK];
  __shared__ float sStart[KK];
  __shared__ float sEnd[KK];
  __shared__ int sTags[TT];
  __shared__ float red[256];
  __shared__ int sLen;

  const int tid = threadIdx.x;
  const int b = blockIdx.x;

  for (int i = tid; i < KK * KK; i += 256) {
    int k = i / KK, j = i % KK;
    bool allowed = (j == 0) || ((j & 1) == 1) || (k == j) || (k == j - 1);
    sTrans[i] = allowed ? trans[i] : IMPOSSIBLE_F;
  }
  if (tid < KK) {
    sStart[tid] = ((tid >= 2) && ((tid & 1) == 0)) ? IMPOSSIBLE_F : startT[tid];
    sEnd[tid] = endT[tid];
  }
  const int* tgRow = tags + (size_t)b * TT;
  for (int i = tid; i < TT; i += 256) sTags[i] = tgRow[i];
  if (tid == 0) {
    const int* mkRow = mask + (size_t)b * TT;
    int blo = 0, bhi = TT;
    for (int it = 0; it < 10; ++it) {
      int mid = (blo + bhi) >> 1;
      int mv = mkRow[mid];
      bool active = blo < bhi;
      bool take = active && (mv != 0);
      blo = take ? (mid + 1) : blo;
      bhi = (active && !take) ? mid : bhi;
    }
    sLen = blo;
  }
  __syncthreads();

  const int len = sLen;
  const float* emB = em + (size_t)b * TT * KK;
  float acc = 0.0f;
  for (int t = tid; t < TT; t += 256) {
    int tg = sTags[t];
    float emTag = emB[(size_t)t * KK + tg];
    if (t == 0) {
      acc += sStart[tg] + emTag;
    } else {
      float tr = sTrans[sTags[t - 1] * KK + tg];
      acc += (t < len) ? (tr + emTag) : 0.0f;
    }
  }
  if (tid == 0) {
    int lt = sTags[(len >= 1) ? (len - 1) : 0];
    acc += sEnd[lt];
  }

  red[tid] = acc;
  __syncthreads();
  for (int k = 128; k > 0; k >>= 1) {
    if (tid < k) red[tid] += red[tid + k];
    __syncthreads();
  }
  if (tid == 0) znll[b] = red[0] - znll[b];  // nll[b] = score - z (in place)
}

// ========================= Kernel 3: mean ==================================
__global__ void __launch_bounds__(1024)
crf_reduce(const float* __restrict__ nll, float* __restrict__ out) {
  __shared__ float s[1024];
  int i = threadIdx.x;
  s[i] = nll[i] + nll[i + 1024] + nll[i + 2048] + nll[i + 3072];
  __syncthreads();
  for (int k = 512; k > 0; k >>= 1) {
    if (i < k) s[i] += s[i + k];
    __syncthreads();
  }
  if (i == 0) out[0] = s[0] * (1.0f / (float)BBATCH);
}

// ===========================================================================
extern "C" void kernel_launch(void* const* d_in, const int* in_sizes, int n_in,
                              void* d_out, int out_size, void* d_ws, size_t ws_size,
                              hipStream_t stream) {
  const float* emissions = (const float*)d_in[0];
  const int* mask = (const int*)d_in[1];       // bool mask delivered as int32
  const int* tags = (const int*)d_in[2];
  const float* trans = (const float*)d_in[3];
  const float* startT = (const float*)d_in[4];
  const float* endT = (const float*)d_in[5];
  float* out = (float*)d_out;
  float* znll = (float*)d_ws;                  // 4096 floats: z, then nll

  crf_scan<<<BBATCH / 16, 32, 0, stream>>>(emissions, mask, trans, startT,
                                           endT, znll);
  crf_score<<<BBATCH, 256, 0, stream>>>(emissions, mask, tags, trans, startT,
                                        endT, znll);
  crf_reduce<<<1, 1024, 0, stream>>>(znll, out);
}